// WaveletTokenizer_8435315769489
// MI455X (gfx1250) — compile-verified
//
#include <hip/hip_runtime.h>
#include <hip/hip_bf16.h>
#include <stdint.h>

// ---------------- constants ----------------
#define INV_SQRT2F 0.70710678118654752440f
#define PAD_TOK    2
#define SLOTS      37
#define NB_PER_CH  4096            // (256/4)*(256/4)
#define L_PER_B    1818624         // 3 * 4 * 4096 * 37
#define NCHUNK     444             // L_PER_B / 4096 (exact)
#define NIMG       48              // 16 * 3
#define NBATCH     16
#define IMG_ELEMS  65536           // 256*256

typedef __attribute__((ext_vector_type(2))) float v2f;
typedef __attribute__((ext_vector_type(8))) float v8f;

// Single-level Haar analysis matrix element W1_s[r][c] (pywt 'haar' normalization).
__device__ __forceinline__ float w1val(int s, int r, int c) {
    int h = s >> 1;
    if (r < h) {
        return ((c >> 1) == r) ? INV_SQRT2F : 0.0f;
    }
    int j = r - h;
    if ((c >> 1) == j) return (c & 1) ? -INV_SQRT2F : INV_SQRT2F;
    return 0.0f;
}

// ---------------- DWT via WMMA (fp32 16x16x4) ----------------
// Y[s x s] = W1_s * X[s x s]   (per image; both strided 256)
__global__ void k_dwt_rows(const float* __restrict__ src, float* __restrict__ Y, int s) {
    int img  = blockIdx.y;
    int tpr  = s >> 4;
    int tr   = (blockIdx.x / tpr) * 16;
    int tc   = (blockIdx.x % tpr) * 16;
    int lane = threadIdx.x;
    int m    = lane & 15;
    int half = lane >> 4;

    const float* X  = src + (size_t)img * IMG_ELEMS;
    float*       Yp = Y   + (size_t)img * IMG_ELEMS;

    int hS = s >> 1;
    int jb = (tr < hS) ? tr : (tr - hS);
    int kstart = 2 * jb;   // rows of this tile only touch K in [kstart, kstart+32)

    v8f acc = {0.f, 0.f, 0.f, 0.f, 0.f, 0.f, 0.f, 0.f};
    #pragma unroll
    for (int kk = 0; kk < 32; kk += 4) {
        int kb = kstart + kk + half * 2;
        v2f a, b;
        a.x = w1val(s, tr + m, kb);
        a.y = w1val(s, tr + m, kb + 1);
        b.x = X[(size_t)(kb)     * 256 + tc + m];
        b.y = X[(size_t)(kb + 1) * 256 + tc + m];
        acc = __builtin_amdgcn_wmma_f32_16x16x4_f32(false, a, false, b,
                                                    (short)0, acc, false, false);
    }
    #pragma unroll
    for (int v = 0; v < 8; v++)
        Yp[(size_t)(tr + v + half * 8) * 256 + tc + m] = acc[v];
}

// P[s x s] = Y[s x s] * W1_s^T  (written into top-left of packed buffer)
__global__ void k_dwt_cols(const float* __restrict__ Y, float* __restrict__ P, int s) {
    int img  = blockIdx.y;
    int tpr  = s >> 4;
    int tr   = (blockIdx.x / tpr) * 16;
    int tc   = (blockIdx.x % tpr) * 16;
    int lane = threadIdx.x;
    int m    = lane & 15;
    int half = lane >> 4;

    const float* Yp = Y + (size_t)img * IMG_ELEMS;
    float*       Pp = P + (size_t)img * IMG_ELEMS;

    int hS = s >> 1;
    int jb = (tc < hS) ? tc : (tc - hS);
    int kstart = 2 * jb;   // cols of this tile only touch K in [kstart, kstart+32)

    v8f acc = {0.f, 0.f, 0.f, 0.f, 0.f, 0.f, 0.f, 0.f};
    #pragma unroll
    for (int kk = 0; kk < 32; kk += 4) {
        int kb = kstart + kk + half * 2;
        v2f a, b;
        a.x = Yp[(size_t)(tr + m) * 256 + kb];
        a.y = Yp[(size_t)(tr + m) * 256 + kb + 1];
        b.x = w1val(s, tc + m, kb);       // B[k][n] = W1^T[k][n] = W1[n][k]
        b.y = w1val(s, tc + m, kb + 1);
        acc = __builtin_amdgcn_wmma_f32_16x16x4_f32(false, a, false, b,
                                                    (short)0, acc, false, false);
    }
    #pragma unroll
    for (int v = 0; v < 8; v++)
        Pp[(size_t)(tr + v + half * 8) * 256 + tc + m] = acc[v];
}

// ---------------- per-image max|coeff| ----------------
__global__ void k_maxabs(const float* __restrict__ P, float* __restrict__ mx) {
    int img = blockIdx.x;
    int t   = threadIdx.x;
    const float* p = P + (size_t)img * IMG_ELEMS;
    float m = 0.f;
    for (int i = t; i < IMG_ELEMS; i += 256) m = fmaxf(m, fabsf(p[i]));
    __shared__ float sh[256];
    sh[t] = m; __syncthreads();
    for (int o = 128; o > 0; o >>= 1) { if (t < o) sh[t] = fmaxf(sh[t], sh[t + o]); __syncthreads(); }
    if (t == 0) mx[img] = sh[0];
}

// ---------------- bitplane tokenizer ----------------
// One thread per 4x4 block; emits 4 bitplanes x 37 slots, token | 0x80 if valid.
__global__ void k_tokenize(const float* __restrict__ P, const float* __restrict__ mx,
                           unsigned char* __restrict__ TV) {
    int t = blockIdx.x * blockDim.x + threadIdx.x;   // < 16*3*4096 = 196608
    int b      = t / (3 * NB_PER_CH);
    int rem    = t % (3 * NB_PER_CH);
    int c      = rem / NB_PER_CH;
    int blkIdx = rem % NB_PER_CH;
    int img    = b * 3 + c;
    int bi = blkIdx >> 6, bj = blkIdx & 63;

    const float* base = P + (size_t)img * IMG_ELEMS + (size_t)(bi * 4) * 256 + bj * 4;

    // load into xs order: index = s4*4 + p, s4=(si,sj), p=(iii,jjj)
    float xx[16], approx[16];
    #pragma unroll
    for (int s4 = 0; s4 < 4; s4++) {
        int si = s4 >> 1, sj = s4 & 1;
        #pragma unroll
        for (int p = 0; p < 4; p++) {
            int iii = p >> 1, jjj = p & 1;
            xx[s4 * 4 + p] = base[(2 * si + iii) * 256 + (2 * sj + jjj)];
            approx[s4 * 4 + p] = 0.f;
        }
    }

    float ma = mx[img];
    float mt = (ma > 0.f) ? ceilf(log2f(ma)) : 0.f;
    float T  = exp2f(mt - 1.f);   // T0

    for (int k = 0; k < 4; k++) {
        bool active = (k == 0) || (T >= 0.125f);
        float twoT = 2.f * T, t15 = 1.5f * T, qT = 0.25f * T;

        bool sub[4]; bool blkSig = false;
        #pragma unroll
        for (int s4 = 0; s4 < 4; s4++) {
            bool a = false;
            #pragma unroll
            for (int p = 0; p < 4; p++) a |= (fabsf(xx[s4 * 4 + p]) >= T);
            sub[s4] = a; blkSig |= a;
        }

        unsigned char tv[SLOTS];
        tv[0] = (unsigned char)((!blkSig && active) ? 0x80 : 0x00);   // tok 0 (Group4x4)
        #pragma unroll
        for (int s4 = 0; s4 < 4; s4++) {
            int o = 1 + 9 * s4;
            bool ca = blkSig && sub[s4] && active;
            tv[o] = (unsigned char)(1 | ((blkSig && !sub[s4] && active) ? 0x80 : 0)); // Group2x2
            #pragma unroll
            for (int p = 0; p < 4; p++) {
                int i = s4 * 4 + p;
                float xi = xx[i], ax = fabsf(xi);
                bool sigc = (ax >= T);
                bool cA = (ax >= twoT);
                bool cB = sigc && !cA;
                int tok2 = (ax > t15) ? 6 : 5;
                int tok1;
                if (cA)      tok1 = (ax > fabsf(approx[i])) ? 6 : 5;
                else if (cB) tok1 = (xi > 0.f) ? 3 : 4;
                else         tok1 = PAD_TOK;
                tv[o + 1 + 2 * p] = (unsigned char)(tok1 | (ca ? 0x80 : 0));
                tv[o + 2 + 2 * p] = (unsigned char)(tok2 | ((ca && cB) ? 0x80 : 0));
                if (ca) {
                    float sgn = (xi > 0.f) ? 1.f : ((xi < 0.f) ? -1.f : 0.f);
                    if (cA)      approx[i] += sgn * ((tok1 == 6) ? qT : -qT);
                    else if (cB) approx[i]  = sgn * (t15 + ((tok2 == 6) ? qT : -qT));
                }
            }
        }

        size_t ob = (size_t)b * L_PER_B + ((size_t)(((c * 4 + k) << 12) + blkIdx)) * SLOTS;
        #pragma unroll
        for (int q = 0; q < SLOTS; q++) TV[ob + q] = tv[q];

        T *= 0.5f;
    }
}

// ---------------- compaction ----------------
__global__ void k_fill(int4* __restrict__ out, long n4) {
    long i = (long)blockIdx.x * blockDim.x + threadIdx.x;
    if (i < n4) out[i] = make_int4(PAD_TOK, PAD_TOK, PAD_TOK, PAD_TOK);
}

__global__ void k_chunksum(const unsigned char* __restrict__ TV, int* __restrict__ cS, int nchunk) {
    int cid = blockIdx.x, b = blockIdx.y, t = threadIdx.x;
    const unsigned char* p = TV + (size_t)b * L_PER_B + (size_t)cid * 4096 + (size_t)t * 16;
    uint4 w = *(const uint4*)p;
    int cnt = __popc(w.x & 0x80808080u) + __popc(w.y & 0x80808080u)
            + __popc(w.z & 0x80808080u) + __popc(w.w & 0x80808080u);
    __shared__ int sh[256];
    sh[t] = cnt; __syncthreads();
    for (int o = 128; o > 0; o >>= 1) { if (t < o) sh[t] += sh[t + o]; __syncthreads(); }
    if (t == 0) cS[b * nchunk + cid] = sh[0];
}

__global__ void k_scanchunks(const int* __restrict__ cS, int* __restrict__ cO, int nchunk) {
    int b = blockIdx.x, t = threadIdx.x;              // blockDim = 512 >= nchunk
    __shared__ int sh[512];
    int v = (t < nchunk) ? cS[b * nchunk + t] : 0;
    sh[t] = v; __syncthreads();
    for (int o = 1; o < 512; o <<= 1) {
        int add = (t >= o) ? sh[t - o] : 0;
        __syncthreads();
        sh[t] += add;
        __syncthreads();
    }
    if (t < nchunk) cO[b * nchunk + t] = sh[t] - v;   // exclusive
}

__global__ void k_scatter(const unsigned char* __restrict__ TV, const int* __restrict__ cO,
                          int* __restrict__ out, int nchunk) {
    int cid = blockIdx.x, b = blockIdx.y, t = threadIdx.x;
    const unsigned char* p = TV + (size_t)b * L_PER_B + (size_t)cid * 4096 + (size_t)t * 16;
    uint4 w = *(const uint4*)p;
    unsigned char by[16];
    *(uint4*)by = w;
    int cnt = 0;
    #pragma unroll
    for (int i = 0; i < 16; i++) cnt += (by[i] >> 7);

    __shared__ int sh[256];
    sh[t] = cnt; __syncthreads();
    for (int o = 1; o < 256; o <<= 1) {
        int add = (t >= o) ? sh[t - o] : 0;
        __syncthreads();
        sh[t] += add;
        __syncthreads();
    }
    int pos = cO[b * nchunk + cid] + (sh[t] - cnt);   // exclusive within chunk + chunk offset
    int* ob = out + (size_t)b * L_PER_B;
    #pragma unroll
    for (int i = 0; i < 16; i++) {
        if (by[i] & 0x80) ob[pos++] = (int)(by[i] & 0x7f);
    }
}

// ---------------- launcher ----------------
extern "C" void kernel_launch(void* const* d_in, const int* in_sizes, int n_in,
                              void* d_out, int out_size, void* d_ws, size_t ws_size,
                              hipStream_t stream) {
    (void)in_sizes; (void)n_in; (void)out_size; (void)ws_size;

    const float* img = (const float*)d_in[0];          // [16,3,256,256] fp32
    int* out = (int*)d_out;                             // [16, L_PER_B] int32
    char* ws = (char*)d_ws;

    // workspace layout (bytes)
    float*         P  = (float*)(ws);                                  // 12,582,912
    float*         Yb = (float*)(ws + 12582912);                       // 12,582,912
    unsigned char* TV = (unsigned char*)(ws + 25165824);               // 29,097,984
    float*         mx = (float*)(ws + 54263808);                       // 192 (pad 256)
    int*           cS = (int*)(ws + 54264064);                         // 28,416
    int*           cO = (int*)(ws + 54292480);                         // 28,416

    // 4-level Haar wavedec2: Z = W1_s * LL * W1_s^T per level, packed in place
    for (int level = 0; level < 4; level++) {
        int s = 256 >> level;
        int tiles = (s / 16) * (s / 16);
        const float* src = (level == 0) ? img : (const float*)P;
        k_dwt_rows<<<dim3(tiles, NIMG), 32, 0, stream>>>(src, Yb, s);
        k_dwt_cols<<<dim3(tiles, NIMG), 32, 0, stream>>>(Yb, P, s);
    }

    k_maxabs<<<NIMG, 256, 0, stream>>>(P, mx);
    k_tokenize<<<(NBATCH * 3 * NB_PER_CH) / 256, 256, 0, stream>>>(P, mx, TV);

    long n4 = (long)NBATCH * L_PER_B / 4;               // 7,274,496 int4's
    k_fill<<<(unsigned)((n4 + 255) / 256), 256, 0, stream>>>((int4*)out, n4);

    k_chunksum<<<dim3(NCHUNK, NBATCH), 256, 0, stream>>>(TV, cS, NCHUNK);
    k_scanchunks<<<NBATCH, 512, 0, stream>>>(cS, cO, NCHUNK);
    k_scatter<<<dim3(NCHUNK, NBATCH), 256, 0, stream>>>(TV, cO, out, NCHUNK);
}